// GatedTransformerXL_48103633715847
// MI455X (gfx1250) — compile-verified
//
#include <hip/hip_runtime.h>
#include <math.h>
#include <stdint.h>

// ---------------------------------------------------------------------------
// Gated Transformer-XL forward for MI455X (gfx1250), fp32 end-to-end.
// All GEMMs run on V_WMMA_F32_16X16X4_F32 (exact fp32 semantics; the network
// is HBM-bound on ~420MB of weights, so low-precision WMMA buys nothing).
// Tile loads use GLOBAL_LOAD_ASYNC_TO_LDS_B128 (ASYNCcnt) with a
// double-buffered, branch-free LDS pipeline: issue stage k+1, s_wait_asynccnt,
// barrier, WMMA on stage k; last stage peeled. Workspace: ~976 MB.
// ---------------------------------------------------------------------------

typedef __attribute__((ext_vector_type(2))) float v2f;
typedef __attribute__((ext_vector_type(8))) float v8f;

__device__ __forceinline__ float gelu_f(float x) {
    return 0.5f * x * (1.0f + erff(x * 0.7071067811865475f));
}
__device__ __forceinline__ float sigmoid_f(float x) {
    return 1.0f / (1.0f + expf(-x));
}

// Async 16B copy global -> LDS, tracked by ASYNCcnt. The LDS address operand
// is the low 32 bits of the generic pointer.
__device__ __forceinline__ void async_b128(float* lds, const float* g) {
    asm volatile("global_load_async_to_lds_b128 %0, %1, off"
                 :: "v"((unsigned)(uintptr_t)lds),
                    "v"((unsigned long long)(uintptr_t)g)
                 : "memory");
}

// ---------------------------------------------------------------------------
// Strided-batched GEMM:  C[z] = act( A[z] (MxK) * op(B[z]) + bias )
//   TRANSB=1 : B stored row-major (N,K)  -> C = A @ B^T   (weights, AC/BD)
//   TRANSB=0 : B stored row-major (K,N)  -> C = A @ B     (attn @ V)
// Block tile 64x64, BK=16, double-buffered. 8 waves (wave32): wave (wm,wn) in
// a 2x4 grid owns a 32x16 C sub-tile = two 16x16 f32 WMMA accumulators.
// LDS layouts are async-copy friendly (contiguous 16B chunks, no transpose):
//   As[m][k] (64x16), Bs[n][k] if TRANSB else Bs[k][n].
// All problem dims here are multiples of 64/16 -> no edge handling.
// ---------------------------------------------------------------------------
template <int TRANSB, int ACT>
__global__ __launch_bounds__(256) void gemm_f32_wmma(
    const float* __restrict__ A, const float* __restrict__ B,
    const float* __restrict__ bias, float* __restrict__ C,
    int K, int lda, int ldb, int ldc,
    long long sA, long long sB, long long sC, int bModB)
{
    __shared__ float As[2][64 * 16];
    __shared__ float Bs[2][64 * 16];

    const int tid  = threadIdx.x;
    const int lane = tid & 31;
    const int wave = tid >> 5;
    const int wm   = wave >> 2;          // 0..1 (M direction)
    const int wn   = wave & 3;           // 0..3 (N direction)
    const int z    = blockIdx.z;

    const float* Ab = A + sA * (long long)z;
    const float* Bb = B + sB * (long long)(bModB ? (z % bModB) : z);
    float*       Cb = C + sC * (long long)z;

    const int mblk = blockIdx.y * 64;
    const int nblk = blockIdx.x * 64;

    v8f acc0 = {};
    v8f acc1 = {};

    const int lm    = lane & 15;
    const int half2 = (lane >> 4) << 1;  // 0 or 2 : K offset of upper half-wave

    // Per-thread async-copy coordinates (each thread moves one 16B chunk per
    // tile; one instruction per wave per tile -> ASYNCcnt += 2 per stage).
    const int am = tid >> 2;             // A row 0..63
    const int ak = (tid & 3) << 2;       // A k-chunk 0,4,8,12
    const int bn = TRANSB ? (tid >> 2) : ((tid & 15) << 2);
    const int bk = TRANSB ? ((tid & 3) << 2) : (tid >> 4);

    // Pointer-bump sources: A advances +16 floats per stage; B advances +16
    // (TRANSB: along K within a row) or +16*ldb (16 K-rows) per stage.
    const float* aSrc = Ab + (long long)(mblk + am) * lda + ak;
    const float* bSrc = TRANSB ? (Bb + (long long)(nblk + bn) * ldb + bk)
                               : (Bb + (long long)bk * ldb + (nblk + bn));
    const long long bStep = TRANSB ? 16 : 16 * (long long)ldb;

    float* aDst0 = &As[0][am * 16 + ak];
    float* aDst1 = &As[1][am * 16 + ak];
    float* bDst0 = TRANSB ? &Bs[0][bn * 16 + bk] : &Bs[0][bk * 64 + bn];
    float* bDst1 = TRANSB ? &Bs[1][bn * 16 + bk] : &Bs[1][bk * 64 + bn];

    // Fragment base offsets in LDS
    const int arow0 = (wm * 32 + lm) * 16;        // acc0 rows
    const int arow1 = arow0 + 16 * 16;            // acc1 rows (+16)
    const int brow  = (wn * 16 + lm) * 16;        // TRANSB=1 fragment base
    const int col   = wn * 16 + lm;               // N column within tile

    auto compute = [&](int buf) {
        const float* as = As[buf];
        const float* bs = Bs[buf];
#pragma unroll
        for (int ks = 0; ks < 16; ks += 4) {
            // WMMA f32 16x16x4 (wave32):
            //  A: lane = M row; VGPR j holds K = ks + j + 2*(lane/16)
            //  B: lane = N col; VGPR j holds K = ks + j + 2*(lane/16)
            const int kb = ks + half2;
            v2f a0 = *(const v2f*)&as[arow0 + kb];
            v2f a1 = *(const v2f*)&as[arow1 + kb];
            v2f b;
            if (TRANSB) {
                b = *(const v2f*)&bs[brow + kb];
            } else {
                b[0] = bs[(kb + 0) * 64 + col];
                b[1] = bs[(kb + 1) * 64 + col];
            }
            acc0 = __builtin_amdgcn_wmma_f32_16x16x4_f32(false, a0, false, b,
                                                         (short)0, acc0, false, false);
            acc1 = __builtin_amdgcn_wmma_f32_16x16x4_f32(false, a1, false, b,
                                                         (short)0, acc1, false, false);
        }
    };

    // Prologue: stage 0 in flight.
    const int nk = K >> 4;
    async_b128(aDst0, aSrc); aSrc += 16;
    async_b128(bDst0, bSrc); bSrc += bStep;

    // Steady state: branch-free double-buffered pipeline.
    for (int kt = 0; kt < nk - 1; ++kt) {
        const int buf = kt & 1;
        async_b128(buf ? aDst0 : aDst1, aSrc); aSrc += 16;
        async_b128(buf ? bDst0 : bDst1, bSrc); bSrc += bStep;
        asm volatile("s_wait_asynccnt 2" ::: "memory");   // stage kt landed
        __syncthreads();
        compute(buf);
        __syncthreads();
    }

    // Epilogue: last stage.
    asm volatile("s_wait_asynccnt 0" ::: "memory");
    __syncthreads();
    compute((nk - 1) & 1);

    // C/D layout: lane%16 = N col; VGPR j holds M = j + 8*(lane/16)
    const int   gcol = nblk + col;
    const float bv   = bias ? bias[gcol] : 0.0f;
    const int   hb   = (lane >> 4) << 3;  // 0 or 8
#pragma unroll
    for (int j = 0; j < 8; ++j) {
        const int r0 = mblk + wm * 32 + j + hb;
        float x0 = acc0[j] + bv;
        if (ACT == 1) x0 = gelu_f(x0);
        Cb[(long long)r0 * ldc + gcol] = x0;
        const int r1 = r0 + 16;
        float x1 = acc1[j] + bv;
        if (ACT == 1) x1 = gelu_f(x1);
        Cb[(long long)r1 * ldc + gcol] = x1;
    }
}

// ---------------------------------------------------------------------------
// LayerNorm over rows of 1024 columns. One 256-thread block per row.
// ---------------------------------------------------------------------------
__global__ __launch_bounds__(256) void layernorm_k(
    const float* __restrict__ X, const float* __restrict__ g,
    const float* __restrict__ b, float* __restrict__ Y)
{
    __shared__ float red[256];
    const int tid = threadIdx.x;
    const long long row = blockIdx.x;
    const float4 v = *(const float4*)(X + row * 1024 + tid * 4);

    red[tid] = v.x + v.y + v.z + v.w;
    __syncthreads();
    for (int m = 128; m; m >>= 1) { if (tid < m) red[tid] += red[tid + m]; __syncthreads(); }
    const float mean = red[0] * (1.0f / 1024.0f);
    __syncthreads();

    const float dx = v.x - mean, dy = v.y - mean, dz = v.z - mean, dw = v.w - mean;
    red[tid] = dx * dx + dy * dy + dz * dz + dw * dw;
    __syncthreads();
    for (int m = 128; m; m >>= 1) { if (tid < m) red[tid] += red[tid + m]; __syncthreads(); }
    const float inv = rsqrtf(red[0] * (1.0f / 1024.0f) + 1e-5f);

    const float4 gg = *(const float4*)(g + tid * 4);
    const float4 bb = *(const float4*)(b + tid * 4);
    float4 o;
    o.x = dx * inv * gg.x + bb.x;
    o.y = dy * inv * gg.y + bb.y;
    o.z = dz * inv * gg.z + bb.z;
    o.w = dw * inv * gg.w + bb.w;
    *(float4*)(Y + row * 1024 + tid * 4) = o;
}

// ---------------------------------------------------------------------------
// Fused rel-shift + mask + scale + softmax. One wave per score row (z,i).
// rel_shift(BD)[i,j] == BDraw[i, j + cur-1-i]; the shifted index is in-bounds
// exactly where the causal mask allows (j <= prev+i). Writes P into AC.
// full fixed at 1024 (32 elements / lane).
// ---------------------------------------------------------------------------
__global__ __launch_bounds__(256) void score_softmax_k(
    float* __restrict__ AC, const float* __restrict__ BD,
    int cur, int prev, float scale)
{
    const int lane = threadIdx.x & 31;
    const int wid  = threadIdx.x >> 5;
    const long long row = (long long)blockIdx.x * 8 + wid;  // row = z*cur + i
    const int i = (int)(row % cur);
    float*       ac = AC + row * 1024;
    const float* bd = BD + row * 1024;
    const int shift = cur - 1 - i;
    const int jmax  = prev + i;

    float v[32];
    float mx = -3.0e38f;
#pragma unroll
    for (int t = 0; t < 32; ++t) {
        const int j = t * 32 + lane;
        float s = (j <= jmax) ? (ac[j] + bd[j + shift]) * scale : -1e30f;
        v[t] = s;
        mx = fmaxf(mx, s);
    }
    for (int m = 16; m; m >>= 1) mx = fmaxf(mx, __shfl_xor(mx, m, 32));
    float sum = 0.0f;
#pragma unroll
    for (int t = 0; t < 32; ++t) { v[t] = expf(v[t] - mx); sum += v[t]; }
    for (int m = 16; m; m >>= 1) sum += __shfl_xor(sum, m, 32);
    const float inv = 1.0f / sum;
#pragma unroll
    for (int t = 0; t < 32; ++t) ac[t * 32 + lane] = v[t] * inv;
}

// ---------------------------------------------------------------------------
// Elementwise helpers
// ---------------------------------------------------------------------------
__global__ __launch_bounds__(256) void addrow_k(
    const float* __restrict__ X, const float* __restrict__ v,
    float* __restrict__ Y, long long n, int ncols)
{
    const long long idx = (long long)blockIdx.x * 256 + threadIdx.x;
    if (idx < n) Y[idx] = X[idx] + v[idx % ncols];
}

__global__ __launch_bounds__(256) void gru_pre_k(
    const float* __restrict__ R1, const float* __restrict__ R2,
    const float* __restrict__ x, float* __restrict__ rx, long long n)
{
    const long long idx = (long long)blockIdx.x * 256 + threadIdx.x;
    if (idx < n) {
        const float r = sigmoid_f(R1[idx] + R2[idx]);
        rx[idx] = r * x[idx];
    }
}

__global__ __launch_bounds__(256) void gru_post_k(
    const float* __restrict__ x,
    const float* __restrict__ Z1, const float* __restrict__ Z2,
    const float* __restrict__ G1, const float* __restrict__ G2,
    float* __restrict__ out, long long n)
{
    const long long idx = (long long)blockIdx.x * 256 + threadIdx.x;
    if (idx < n) {
        const float z  = sigmoid_f(Z1[idx] + Z2[idx] - 2.0f);  // GRU_BIAS = 2
        const float hh = tanhf(G1[idx] + G2[idx]);
        out[idx] = (1.0f - z) * x[idx] + z * hh;
    }
}

__global__ __launch_bounds__(256) void pos_k(float* __restrict__ pos, int full, int dim)
{
    const long long idx = (long long)blockIdx.x * 256 + threadIdx.x;
    if (idx >= (long long)full * dim) return;
    const int p = (int)(idx / dim);
    const int c = (int)(idx % dim);
    const int half = dim >> 1;
    const float seqv = (float)(full - 1 - p);
    const int m = (c < half) ? c : (c - half);
    const float inv = expf(-((float)(2 * m) / (float)dim) * 9.210340371976184f); // ln(10000)
    pos[idx] = (c < half) ? sinf(seqv * inv) : cosf(seqv * inv);
}

// h (BS,CUR,IN) -> ht ((i*BS+b), IN)
__global__ __launch_bounds__(256) void tin_k(
    const float* __restrict__ h, float* __restrict__ ht,
    int CURc, int BSc, int INc, long long n4)
{
    const long long idx = (long long)blockIdx.x * 256 + threadIdx.x;
    if (idx >= n4) return;
    const int in4 = INc >> 2;
    const int c4  = (int)(idx % in4);
    const long long r = idx / in4;             // r = i*BS + b
    const int b = (int)(r % BSc);
    const int i = (int)(r / BSc);
    *(float4*)(ht + r * INc + 4 * c4) =
        *(const float4*)(h + ((long long)b * CURc + i) * INc + 4 * c4);
}

// x ((i*BS+b), D) -> y (BS,CUR,D)
__global__ __launch_bounds__(256) void tout_k(
    const float* __restrict__ x, float* __restrict__ y,
    int CURc, int BSc, int Dc, long long n4)
{
    const long long idx = (long long)blockIdx.x * 256 + threadIdx.x;
    if (idx >= n4) return;
    const int d4 = Dc >> 2;
    const int c4 = (int)(idx % d4);
    const long long r = idx / d4;              // r = i*BS + b
    const int b = (int)(r % BSc);
    const int i = (int)(r / BSc);
    *(float4*)(y + ((long long)b * CURc + i) * Dc + 4 * c4) =
        *(const float4*)(x + r * Dc + 4 * c4);
}

// ---------------------------------------------------------------------------
// Host orchestration
// ---------------------------------------------------------------------------
extern "C" void kernel_launch(void* const* d_in, const int* in_sizes, int n_in,
                              void* d_out, int out_size, void* d_ws, size_t ws_size,
                              hipStream_t stream)
{
    (void)in_sizes; (void)n_in; (void)out_size; (void)ws_size;

    const float* h      = (const float*)d_in[0];
    const float* memory = (const float*)d_in[1];
    const float* W_emb  = (const float*)d_in[2];
    const float* b_emb  = (const float*)d_in[3];
    const float* U      = (const float*)d_in[4];
    const float* V      = (const float*)d_in[5];
    const float* ln1    = (const float*)d_in[6];
    const float* ln2    = (const float*)d_in[7];
    const float* Wattn  = (const float*)d_in[8];
    const float* g1     = (const float*)d_in[9];
    const float* g2     = (const float*)d_in[10];
    const float* fcW1   = (const float*)d_in[11];
    const float* fcb1   = (const float*)d_in[12];
    const float* fcW2   = (const float*)d_in[13];
    const float* fcb2   = (const float*)d_in[14];
    float* out_final = (float*)d_out;

    enum { L = 4, D = 1024, NH = 16, HD = 64, CUR = 512, PREV = 512,
           FULL = 1024, BS = 8, IN = 512, FF = 4096 };
    const int MR = CUR * BS;    // 4096 query rows
    const int KR = FULL * BS;   // 8192 key rows
    const long long DD = (long long)D * D;

    float* ws = (float*)d_ws;
    size_t off = 0;
    auto alloc = [&](size_t nf) { float* p = ws + off; off += nf; return p; };
    float* ht  = alloc((size_t)MR * IN);
    float* pos = alloc((size_t)FULL * D);
    float* cur = alloc((size_t)MR * D);         // running activation (cur,bs,D)
    float* key = alloc((size_t)KR * D);
    float* nq  = alloc((size_t)MR * D);         // also reused as normed o1
    float* nk  = alloc((size_t)KR * D);
    float* qh  = alloc((size_t)MR * D);
    float* qhU = alloc((size_t)MR * D);
    float* qhV = alloc((size_t)MR * D);
    float* kh  = alloc((size_t)KR * D);
    float* vh  = alloc((size_t)KR * D);
    float* rh  = alloc((size_t)FULL * D);
    float* AC  = alloc((size_t)BS * NH * CUR * FULL);   // 256 MB (scores -> P)
    float* BD  = alloc((size_t)BS * NH * CUR * FULL);   // 256 MB
    float* Yh  = alloc((size_t)MR * D);
    float* Yo  = alloc((size_t)MR * D);
    float* R1  = alloc((size_t)MR * D);
    float* R2  = alloc((size_t)MR * D);
    float* Z1  = alloc((size_t)MR * D);
    float* Z2  = alloc((size_t)MR * D);
    float* G1  = alloc((size_t)MR * D);
    float* G2  = alloc((size_t)MR * D);
    float* rx  = alloc((size_t)MR * D);
    float* o1  = alloc((size_t)MR * D);
    float* F1  = alloc((size_t)MR * FF);
    float* E2  = alloc((size_t)MR * D);

    auto gemm = [&](const float* A, const float* B, const float* bias, float* C,
                    int M, int N, int K, int lda, int ldb, int ldc,
                    long long sA, long long sB, long long sC, int batches,
                    int transB, int act, int bModB) {
        dim3 grid(N / 64, M / 64, batches);
        if (transB) {
            if (act)
                gemm_f32_wmma<1, 1><<<grid, dim3(256), 0, stream>>>(
                    A, B, bias, C, K, lda, ldb, ldc, sA, sB, sC, bModB);
            else
                gemm_f32_wmma<1, 0><<<grid, dim3(256), 0, stream>>>(
                    A, B, bias, C, K, lda, ldb, ldc, sA, sB, sC, bModB);
        } else {
            gemm_f32_wmma<0, 0><<<grid, dim3(256), 0, stream>>>(
                A, B, bias, C, K, lda, ldb, ldc, sA, sB, sC, bModB);
        }
    };
    auto ew = [&](long long n) { return dim3((unsigned)((n + 255) / 256)); };

    // ---- embedding: x = gelu(h^T @ W_emb^T + b_emb) ----
    tin_k<<<ew((long long)MR * IN / 4), 256, 0, stream>>>(h, ht, CUR, BS, IN,
                                                          (long long)MR * IN / 4);
    pos_k<<<ew((long long)FULL * D), 256, 0, stream>>>(pos, FULL, D);
    gemm(ht, W_emb, b_emb, cur, MR, D, IN, IN, IN, D, 0, 0, 0, 1, 1, 1, 0);

    const long long nMD = (long long)MR * D;
    for (int l = 0; l < L; ++l) {
        const float* Wq  = Wattn + ((size_t)l * 5 + 0) * DD;
        const float* Wk  = Wattn + ((size_t)l * 5 + 1) * DD;
        const float* Wv  = Wattn + ((size_t)l * 5 + 2) * DD;
        const float* Wr_ = Wattn + ((size_t)l * 5 + 3) * DD;
        const float* Wo  = Wattn + ((size_t)l * 5 + 4) * DD;
        const float* g1l = g1 + (size_t)l * 6 * DD;
        const float* g2l = g2 + (size_t)l * 6 * DD;

        // key = concat(mem[l], cur)
        hipMemcpyAsync(key, memory + (size_t)l * PREV * BS * D,
                       sizeof(float) * (size_t)PREV * BS * D,
                       hipMemcpyDeviceToDevice, stream);
        hipMemcpyAsync(key + (size_t)PREV * BS * D, cur,
                       sizeof(float) * (size_t)CUR * BS * D,
                       hipMemcpyDeviceToDevice, stream);

        layernorm_k<<<MR, 256, 0, stream>>>(cur, ln1 + (2 * l) * D, ln1 + (2 * l + 1) * D, nq);
        layernorm_k<<<KR, 256, 0, stream>>>(key, ln1 + (2 * l) * D, ln1 + (2 * l + 1) * D, nk);

        // projections
        gemm(nq,  Wq,  nullptr, qh, MR,   D, D, D, D, D, 0, 0, 0, 1, 1, 0, 0);
        gemm(nk,  Wk,  nullptr, kh, KR,   D, D, D, D, D, 0, 0, 0, 1, 1, 0, 0);
        gemm(nk,  Wv,  nullptr, vh, KR,   D, D, D, D, D, 0, 0, 0, 1, 1, 0, 0);
        gemm(pos, Wr_, nullptr, rh, FULL, D, D, D, D, D, 0, 0, 0, 1, 1, 0, 0);

        addrow_k<<<ew(nMD), 256, 0, stream>>>(qh, U, qhU, nMD, D);
        addrow_k<<<ew(nMD), 256, 0, stream>>>(qh, V, qhV, nMD, D);

        // AC[z=(b,h)] = (q+U) @ K^T   (batch offset b*D + h*HD == z*HD)
        gemm(qhU, kh, nullptr, AC, CUR, FULL, HD, BS * D, BS * D, FULL,
             HD, HD, (long long)CUR * FULL, BS * NH, 1, 0, 0);
        // BDraw[z] = (q+V) @ R^T   (R shared over batch b: offset (z%NH)*HD)
        gemm(qhV, rh, nullptr, BD, CUR, FULL, HD, BS * D, D, FULL,
             HD, HD, (long long)CUR * FULL, BS * NH, 1, 0, NH);

        score_softmax_k<<<BS * NH * CUR / 8, 256, 0, stream>>>(AC, BD, CUR, PREV, 0.125f);

        // Y = P @ V (per head), written head-major into (i*BS+b, D)
        gemm(AC, vh, nullptr, Yh, CUR, HD, FULL, FULL, BS * D, BS * D,
             (long long)CUR * FULL, HD, HD, BS * NH, 0, 0, 0);
        // output projection + GELU
        gemm(Yh, Wo, nullptr, Yo, MR, D, D, D, D, D, 0, 0, 0, 1, 1, 1, 0);

        // ---- GRU gate 1: x=cur, y=Yo ----
        gemm(Yo,  g1l + 0 * DD, nullptr, R1, MR, D, D, D, D, D, 0, 0, 0, 1, 1, 0, 0);
        gemm(cur, g1l + 1 * DD, nullptr, R2, MR, D, D, D, D, D, 0, 0, 0, 1, 1, 0, 0);
        gru_pre_k<<<ew(nMD), 256, 0, stream>>>(R1, R2, cur, rx, nMD);
        gemm(rx,  g1l + 5 * DD, nullptr, G2, MR, D, D, D, D, D, 0, 0, 0, 1, 1, 0, 0);
        gemm(Yo,  g1l + 2 * DD, nullptr, Z1, MR, D, D, D, D, D, 0, 0, 0, 1, 1, 0, 0);
        gemm(cur, g1l + 3 * DD, nullptr, Z2, MR, D, D, D, D, D, 0, 0, 0, 1, 1, 0, 0);
        gemm(Yo,  g1l + 4 * DD, nullptr, G1, MR, D, D, D, D, D, 0, 0, 0, 1, 1, 0, 0);
        gru_post_k<<<ew(nMD), 256, 0, stream>>>(cur, Z1, Z2, G1, G2, o1, nMD);

        // ---- FFN: E = gelu(gelu(LN(o1) W1^T + b1) W2^T + b2) ----
        layernorm_k<<<MR, 256, 0, stream>>>(o1, ln2 + (2 * l) * D, ln2 + (2 * l + 1) * D, nq);
        gemm(nq, fcW1 + (size_t)l * FF * D, fcb1 + (size_t)l * FF, F1,
             MR, FF, D, D, D, FF, 0, 0, 0, 1, 1, 1, 0);
        gemm(F1, fcW2 + (size_t)l * D * FF, fcb2 + (size_t)l * D, E2,
             MR, D, FF, FF, FF, D, 0, 0, 0, 1, 1, 1, 0);

        // ---- GRU gate 2: x=o1, y=E2 -> cur ----
        gemm(E2, g2l + 0 * DD, nullptr, R1, MR, D, D, D, D, D, 0, 0, 0, 1, 1, 0, 0);
        gemm(o1, g2l + 1 * DD, nullptr, R2, MR, D, D, D, D, D, 0, 0, 0, 1, 1, 0, 0);
        gru_pre_k<<<ew(nMD), 256, 0, stream>>>(R1, R2, o1, rx, nMD);
        gemm(rx, g2l + 5 * DD, nullptr, G2, MR, D, D, D, D, D, 0, 0, 0, 1, 1, 0, 0);
        gemm(E2, g2l + 2 * DD, nullptr, Z1, MR, D, D, D, D, D, 0, 0, 0, 1, 1, 0, 0);
        gemm(o1, g2l + 3 * DD, nullptr, Z2, MR, D, D, D, D, D, 0, 0, 0, 1, 1, 0, 0);
        gemm(E2, g2l + 4 * DD, nullptr, G1, MR, D, D, D, D, D, 0, 0, 0, 1, 1, 0, 0);
        gru_post_k<<<ew(nMD), 256, 0, stream>>>(o1, Z1, Z2, G1, G2, cur, nMD);
    }

    // (cur,bs,D) -> (bs,cur,D)
    tout_k<<<ew(nMD / 4), 256, 0, stream>>>(cur, out_final, CUR, BS, D, nMD / 4);
}